// Voxelization_18245021073712
// MI455X (gfx1250) — compile-verified
//
#include <hip/hip_runtime.h>

// Shapes fixed by the reference: B=16, C=64, N=32768, R=32, R^3 = 32768.
#define NB 16
#define NC 64
#define NN 32768           // points per batch (2^15)
#define RR 32
#define R3 (RR*RR*RR)      // 32768 = 2^15

typedef float v2f __attribute__((ext_vector_type(2)));
typedef float v8f __attribute__((ext_vector_type(8)));

// ---------------------------------------------------------------------------
// Zero a buffer with 16B stores (harness poisons d_out/d_ws; sums/counters
// must start at zero on every call).
// ---------------------------------------------------------------------------
__global__ __launch_bounds__(256) void vx_zero_f4(float4* __restrict__ p, int n4) {
  int i = blockIdx.x * blockDim.x + threadIdx.x;
  int stride = gridDim.x * blockDim.x;
  float4 z; z.x = 0.f; z.y = 0.f; z.z = 0.f; z.w = 0.f;
  for (; i < n4; i += stride) p[i] = z;
}

// ---------------------------------------------------------------------------
// Per-batch, per-axis coordinate mean via V_WMMA_F32_16X16X4_F32 (exact f32
// FMA).  A = 64 consecutive coords in an arbitrary bijective 16x4 layout
// (each lane loads a float2), B = ones(4x16) => D[i][j] += rowsum_i, so
// sum(all 256 D entries) = 16 * sum(A).  mean = total / (16*N), a power-of-
// two scale (exact).  Grid = B*3 blocks of 1024 threads (32 waves); each wave
// covers 16 chunks with 4 independent accumulator chains for latency hiding.
// ---------------------------------------------------------------------------
__global__ __launch_bounds__(1024) void vx_mean_wmma(const float* __restrict__ coords,
                                                     float* __restrict__ mean) {
  __shared__ float sdata[1024];
  const int b    = blockIdx.x / 3;
  const int a    = blockIdx.x % 3;
  const int tid  = threadIdx.x;
  const int lane = tid & 31;     // wave32
  const int wave = tid >> 5;     // 32 waves per block

  v2f ones; ones.x = 1.0f; ones.y = 1.0f;

  // Each wave owns 16 chunks of 64 floats = 1024 consecutive floats.
  const float* q = coords + ((size_t)b * 3 + a) * NN + wave * 1024 + lane * 2;

  v8f a0 = {}, a1 = {}, a2 = {}, a3 = {};
#pragma unroll
  for (int it = 0; it < 4; ++it) {
    v2f x0 = *(const v2f*)(q + it * 256 + 0);
    v2f x1 = *(const v2f*)(q + it * 256 + 64);
    v2f x2 = *(const v2f*)(q + it * 256 + 128);
    v2f x3 = *(const v2f*)(q + it * 256 + 192);
    a0 = __builtin_amdgcn_wmma_f32_16x16x4_f32(false, x0, false, ones, (short)0, a0, false, false);
    a1 = __builtin_amdgcn_wmma_f32_16x16x4_f32(false, x1, false, ones, (short)0, a1, false, false);
    a2 = __builtin_amdgcn_wmma_f32_16x16x4_f32(false, x2, false, ones, (short)0, a2, false, false);
    a3 = __builtin_amdgcn_wmma_f32_16x16x4_f32(false, x3, false, ones, (short)0, a3, false, false);
  }
  float s = 0.f;
#pragma unroll
  for (int k = 0; k < 8; ++k) s += a0[k] + a1[k] + a2[k] + a3[k];

  sdata[tid] = s;
  __syncthreads();
  for (int off = 512; off > 0; off >>= 1) {
    if (tid < off) sdata[tid] += sdata[tid + off];
    __syncthreads();
  }
  if (tid == 0) mean[b * 3 + a] = sdata[0] * (1.0f / (16.0f * (float)NN));
}

// ---------------------------------------------------------------------------
// Per-batch max ||c - mean||^2.  One point per thread (2048 blocks -> one
// memory round-trip of latency), block LDS max-reduce, then a u32 atomicMax
// on the float bit pattern (values >= 0, so uint order == float order).
// Max is exact & commutative => deterministic despite atomics.
// ---------------------------------------------------------------------------
__global__ __launch_bounds__(256) void vx_maxred(const float* __restrict__ coords,
                                                 const float* __restrict__ mean,
                                                 unsigned* __restrict__ maxbits) {
  __shared__ float sdata[256];
  const int pid = blockIdx.x * 256 + threadIdx.x;     // [0, NB*NN)
  const int b = pid >> 15;
  const int n = pid & (NN - 1);
  const float* c0 = coords + (size_t)b * 3 * NN;
  const float dx = c0[n]          - mean[b * 3 + 0];
  const float dy = c0[NN + n]     - mean[b * 3 + 1];
  const float dz = c0[2 * NN + n] - mean[b * 3 + 2];
  sdata[threadIdx.x] = dx * dx + dy * dy + dz * dz;
  __syncthreads();
  for (int off = 128; off > 0; off >>= 1) {
    if (threadIdx.x < off)
      sdata[threadIdx.x] = fmaxf(sdata[threadIdx.x], sdata[threadIdx.x + off]);
    __syncthreads();
  }
  if (threadIdx.x == 0) atomicMax(&maxbits[b], __float_as_uint(sdata[0]));
}

// ---------------------------------------------------------------------------
// Fused: norm_coords (output #2) + voxel index + atomic scatter of features
// and counts.  rintf = round-half-to-even, matching jnp.round.  True f32
// division to track the reference.  Feature reads are fully coalesced and
// non-temporal (streamed once, 128 MB) so the 134 MB atomic accumulation
// surface stays resident in the 192 MB L2.
// ---------------------------------------------------------------------------
__global__ __launch_bounds__(256) void vx_scatter(const float* __restrict__ feat,
                                                  const float* __restrict__ coords,
                                                  const float* __restrict__ mean,
                                                  const unsigned* __restrict__ maxbits,
                                                  float* __restrict__ ncout,
                                                  float* __restrict__ out,
                                                  float* __restrict__ cnt) {
  const int pid = blockIdx.x * 256 + threadIdx.x;
  const int b = pid >> 15;
  const int n = pid & (NN - 1);
  const float dn = fmaxf(2.0f * sqrtf(__uint_as_float(maxbits[b])), 1e-6f);

  int v3[3];
#pragma unroll
  for (int a = 0; a < 3; ++a) {
    float c  = coords[((size_t)b * 3 + a) * NN + n];
    float nc = (c - mean[b * 3 + a]) / dn + 0.5f;
    float t  = fminf(fmaxf(nc * (float)RR, 0.0f), (float)(RR - 1));
    __builtin_nontemporal_store(t, &ncout[((size_t)b * 3 + a) * NN + n]);
    v3[a] = (int)rintf(t);
  }
  const int v = v3[0] * (RR * RR) + v3[1] * RR + v3[2];

  unsafeAtomicAdd(&cnt[((size_t)b << 15) + v], 1.0f);
  const float* f = feat + (size_t)b * NC * NN + n;
  float* o = out + (size_t)b * NC * R3 + v;
#pragma unroll
  for (int c = 0; c < NC; ++c)
    unsafeAtomicAdd(o + (size_t)c * R3, __builtin_nontemporal_load(f + (size_t)c * NN));
}

// ---------------------------------------------------------------------------
// out[b,c,v] /= max(cnt[b,v], 1)  (empty voxels stay 0, as in the reference).
// ---------------------------------------------------------------------------
__global__ __launch_bounds__(256) void vx_normalize(float* __restrict__ out,
                                                    const float* __restrict__ cnt) {
  const size_t total = (size_t)NB * NC * R3;
  size_t i = (size_t)blockIdx.x * blockDim.x + threadIdx.x;
  const size_t stride = (size_t)gridDim.x * blockDim.x;
  for (; i < total; i += stride) {
    const int v = (int)(i & (R3 - 1));       // R3 = 2^15
    const int b = (int)(i >> 21);            // NC*R3 = 2^21
    const float cc = cnt[((size_t)b << 15) + v];
    out[i] = out[i] / fmaxf(cc, 1.0f);
  }
}

extern "C" void kernel_launch(void* const* d_in, const int* in_sizes, int n_in,
                              void* d_out, int out_size, void* d_ws, size_t ws_size,
                              hipStream_t stream) {
  const float* feat   = (const float*)d_in[0];   // [B, C, N]
  const float* coords = (const float*)d_in[1];   // [B, 3, N]

  float* out   = (float*)d_out;                  // [B, C, 32,32,32]
  float* ncout = out + (size_t)NB * NC * R3;     // [B, 3, N]

  // Workspace (~2.1 MB): mean f32[48] @0 | maxbits u32[16] @64 | cnt f32[B*R3] @128
  float*    ws_f    = (float*)d_ws;
  float*    mean    = ws_f;
  unsigned* maxbits = (unsigned*)(ws_f + 64);
  float*    cnt     = ws_f + 128;

  (void)in_sizes; (void)n_in; (void)out_size; (void)ws_size;

  // Zero: the sum surface in d_out, and ws (mean slot + maxbits + cnt).
  vx_zero_f4<<<4096, 256, 0, stream>>>((float4*)out, (NB * NC * R3) / 4);
  vx_zero_f4<<<512, 256, 0, stream>>>((float4*)ws_f, (128 + NB * R3) / 4);

  vx_mean_wmma<<<NB * 3, 1024, 0, stream>>>(coords, mean);
  vx_maxred<<<(NB * NN) / 256, 256, 0, stream>>>(coords, mean, maxbits);
  vx_scatter<<<(NB * NN) / 256, 256, 0, stream>>>(feat, coords, mean, maxbits,
                                                  ncout, out, cnt);
  vx_normalize<<<4096, 256, 0, stream>>>(out, cnt);
}